// CrossSlideConsistencyLoss_67705864454620
// MI455X (gfx1250) — compile-verified
//
#include <hip/hip_runtime.h>
#include <math.h>

typedef __attribute__((ext_vector_type(2))) float v2f;
typedef __attribute__((ext_vector_type(8))) float v8f;

#define N_PTS   8192
#define NRB     20
#define NAB     72
#define DESC_STRIDE 32   // 25 used, padded
#define CH      2048     // j-chunk staged in LDS (32 KB)

// ---------------------------------------------------------------------------
// Kernel 1: per-slide radial histogram + angular power spectrum (desc[0..24]).
// One block (1024 threads) per slide; three passes over 8K points.
// ---------------------------------------------------------------------------
__global__ __launch_bounds__(1024)
void desc_kernel(const float* __restrict__ c0, const float* __restrict__ c1,
                 const float* __restrict__ c2, float* __restrict__ desc)
{
    const float* coords = (blockIdx.x == 0) ? c0 : (blockIdx.x == 1 ? c1 : c2);
    const float2* pts = (const float2*)coords;
    const int tid = threadIdx.x;

    __shared__ float red[1024];
    __shared__ float s_cx, s_cy, s_rmax;
    __shared__ int   rh[NRB];
    __shared__ int   ah[NAB];
    __shared__ float ahc[NAB];

    // ---- pass 1: centroid ----
    float sx = 0.f, sy = 0.f;
    for (int i = tid; i < N_PTS; i += 1024) { float2 p = pts[i]; sx += p.x; sy += p.y; }
    red[tid] = sx; __syncthreads();
    for (int s = 512; s > 0; s >>= 1) { if (tid < s) red[tid] += red[tid + s]; __syncthreads(); }
    if (tid == 0) s_cx = red[0] * (1.0f / N_PTS);
    __syncthreads();
    red[tid] = sy; __syncthreads();
    for (int s = 512; s > 0; s >>= 1) { if (tid < s) red[tid] += red[tid + s]; __syncthreads(); }
    if (tid == 0) s_cy = red[0] * (1.0f / N_PTS);
    __syncthreads();

    // ---- pass 2: max radius ----
    const float cx = s_cx, cy = s_cy;
    float mx = 0.f;
    for (int i = tid; i < N_PTS; i += 1024) {
        float2 p = pts[i];
        float dx = p.x - cx, dy = p.y - cy;
        mx = fmaxf(mx, sqrtf(dx * dx + dy * dy));
    }
    red[tid] = mx; __syncthreads();
    for (int s = 512; s > 0; s >>= 1) { if (tid < s) red[tid] = fmaxf(red[tid], red[tid + s]); __syncthreads(); }
    if (tid == 0) s_rmax = red[0];
    if (tid < NRB) rh[tid] = 0;
    if (tid < NAB) ah[tid] = 0;
    __syncthreads();

    // ---- pass 3: histograms (LDS atomics) ----
    const float inv_r = 1.0f / (s_rmax + 1e-8f);
    const float PI_F  = 3.14159265358979f;
    for (int i = tid; i < N_PTS; i += 1024) {
        float2 p = pts[i];
        float dx = p.x - cx, dy = p.y - cy;
        float rn = sqrtf(dx * dx + dy * dy) * inv_r;
        int ri = (int)floorf(rn * (float)NRB);
        ri = ri < 0 ? 0 : (ri > NRB - 1 ? NRB - 1 : ri);
        atomicAdd(&rh[ri], 1);
        float ang = atan2f(dy, dx);
        int ai = (int)floorf((ang + PI_F) * ((float)NAB / (2.0f * PI_F)));
        ai = ai < 0 ? 0 : (ai > NAB - 1 ? NAB - 1 : ai);
        atomicAdd(&ah[ai], 1);
    }
    __syncthreads();

    // mean of angular counts is exactly N/72
    if (tid < NAB) ahc[tid] = (float)ah[tid] - (float)N_PTS / (float)NAB;
    if (tid < NRB) desc[blockIdx.x * DESC_STRIDE + tid] = (float)rh[tid] * (1.0f / N_PTS);
    __syncthreads();

    // ---- direct DFT, first 5 magnitudes ----
    if (tid < 5) {
        float cr = 0.f, ci = 0.f;
        const float w = -2.0f * PI_F * (float)tid / (float)NAB;
        for (int k = 0; k < NAB; ++k) {
            float ph = w * (float)k;
            cr += ahc[k] * cosf(ph);
            ci += ahc[k] * sinf(ph);
        }
        desc[blockIdx.x * DESC_STRIDE + NRB + tid] = sqrtf(cr * cr + ci * ci);
    }
}

// ---------------------------------------------------------------------------
// Kernel 2: pairwise min-distance via V_WMMA_F32_16X16X4_F32.
//   A row i = [x_i, y_i, sq_i, 1];  B col j = [-2x_j, -2y_j, 1, sq_j]
//   => D[i][j] = sq_i + sq_j - 2<p_i,p_j> = d2_{ij} in one WMMA.
// B fragments precomputed per 2048-pt chunk into LDS (per-half arrays):
// steady-state tile = ds_load(0.5) + wmma + 4x min3. Diagonal tile peeled;
// i0 is readfirstlane'd so all control flow around the WMMAs is scalar
// (s_cmp/s_cbranch, EXEC provably all-1s as the ISA requires).
// ---------------------------------------------------------------------------
__global__ __launch_bounds__(256)
void collide_kernel(const float* __restrict__ c0, const float* __restrict__ c1,
                    const float* __restrict__ c2, int* __restrict__ counters)
{
    __shared__ float2 sA[CH];   // K=0,1 half:  (-2x_j, -2y_j)
    __shared__ float2 sB[CH];   // K=2,3 half:  (1, sq_j)

    const float* coords = (blockIdx.y == 0) ? c0 : (blockIdx.y == 1 ? c1 : c2);
    const float2* pts = (const float2*)coords;

    const int lane = threadIdx.x & 31;
    const int half = lane >> 4;          // 0: K=0,1 / rows 0-7 ; 1: K=2,3 / rows 8-15
    const int t    = lane & 15;          // column lane / row-of-A index
    const int m_base = half * 8;
    // wave-uniform row-strip base, forced into an SGPR so every branch that
    // depends on it is scalar (keeps EXEC all-1s around the WMMAs).
    const int i0 = __builtin_amdgcn_readfirstlane(
        (blockIdx.x * 8 + (threadIdx.x >> 5)) * 16);

    // A fragment (constant over the whole sweep)
    v2f a;
    {
        float2 p = pts[i0 + t];
        float sq = p.x * p.x + p.y * p.y;
        a[0] = half ? sq   : p.x;
        a[1] = half ? 1.0f : p.y;
    }

    float mn[8];
#pragma unroll
    for (int v = 0; v < 8; ++v) mn[v] = 3.4e38f;

    const float2* __restrict__ src = half ? sB : sA;   // per-lane half select
    const v8f cz = {};

    for (int chunk = 0; chunk < N_PTS; chunk += CH) {
        __syncthreads();                       // previous chunk fully consumed
        for (int k = threadIdx.x; k < CH; k += 256) {
            __builtin_prefetch(&pts[chunk + CH + k], 0, 1);   // next chunk
            float2 p = pts[chunk + k];
            sA[k] = make_float2(-2.0f * p.x, -2.0f * p.y);
            sB[k] = make_float2(1.0f, p.x * p.x + p.y * p.y);
        }
        __syncthreads();

        auto sweep = [&](int lo, int hi) {     // mask-free hot loop
#pragma unroll 4
            for (int jj = lo; jj < hi; jj += 16) {
                float2 q = src[jj + t];
                v2f b; b[0] = q.x; b[1] = q.y;
                v8f d = __builtin_amdgcn_wmma_f32_16x16x4_f32(
                    false, a, false, b, (short)0, cz, false, false);
#pragma unroll
                for (int v = 0; v < 8; ++v) mn[v] = fminf(mn[v], d[v]);
            }
        };

        const int dloc = i0 - chunk;           // scalar (SGPR)
        if (dloc >= 0 && dloc < CH) {
            sweep(0, dloc);
            {   // peeled diagonal tile: mask the self pair only here
                float2 q = src[dloc + t];
                v2f b; b[0] = q.x; b[1] = q.y;
                v8f d = __builtin_amdgcn_wmma_f32_16x16x4_f32(
                    false, a, false, b, (short)0, cz, false, false);
#pragma unroll
                for (int v = 0; v < 8; ++v) {
                    float dv = ((m_base + v) == t) ? 3.4e38f : d[v];
                    mn[v] = fminf(mn[v], dv);
                }
            }
            sweep(dloc + 16, CH);
        } else {
            sweep(0, CH);
        }
    }

    // min-reduce over the 16 column lanes of each half (wave32 shuffles)
#pragma unroll
    for (int v = 0; v < 8; ++v) {
        float m = mn[v];
        m = fminf(m, __shfl_xor(m, 1, 32));
        m = fminf(m, __shfl_xor(m, 2, 32));
        m = fminf(m, __shfl_xor(m, 4, 32));
        m = fminf(m, __shfl_xor(m, 8, 32));
        mn[v] = m;
    }
    if (t == 0) {   // lane 0 -> rows i0..i0+7, lane 16 -> rows i0+8..i0+15
        int cnt = 0;
#pragma unroll
        for (int v = 0; v < 8; ++v) cnt += (mn[v] < 1e-4f) ? 1 : 0;  // d<0.01 <=> d2<1e-4
        if (cnt) atomicAdd(&counters[blockIdx.y], cnt);
    }
}

// ---------------------------------------------------------------------------
// Kernel 3: unbiased variance across the 3 slides, averaged over 26 features.
// ---------------------------------------------------------------------------
__global__ void finalize_kernel(const float* __restrict__ desc,
                                const int* __restrict__ counters,
                                float* __restrict__ out)
{
    if (threadIdx.x == 0 && blockIdx.x == 0) {
        float acc = 0.f;
        for (int f = 0; f < 26; ++f) {
            float v[3];
            for (int s = 0; s < 3; ++s)
                v[s] = (f < 25) ? desc[s * DESC_STRIDE + f]
                                : (float)counters[s] * (1.0f / N_PTS);
            float m = (v[0] + v[1] + v[2]) * (1.0f / 3.0f);
            float var = ((v[0] - m) * (v[0] - m) + (v[1] - m) * (v[1] - m) +
                         (v[2] - m) * (v[2] - m)) * 0.5f;   // ddof=1
            acc += var;
        }
        out[0] = acc * (1.0f / 26.0f);
    }
}

// ---------------------------------------------------------------------------
extern "C" void kernel_launch(void* const* d_in, const int* in_sizes, int n_in,
                              void* d_out, int out_size, void* d_ws, size_t ws_size,
                              hipStream_t stream)
{
    (void)in_sizes; (void)n_in; (void)out_size; (void)ws_size;
    const float* c0 = (const float*)d_in[0];
    const float* c1 = (const float*)d_in[1];
    const float* c2 = (const float*)d_in[2];
    // d_in[3] (slide_labels) unused by the math, as in the reference.

    float* desc     = (float*)d_ws;                                   // 3*32 floats
    int*   counters = (int*)((char*)d_ws + 3 * DESC_STRIDE * sizeof(float));

    hipMemsetAsync(counters, 0, 3 * sizeof(int), stream);             // graph-capture safe

    desc_kernel<<<dim3(3), dim3(1024), 0, stream>>>(c0, c1, c2, desc);
    collide_kernel<<<dim3(64, 3), dim3(256), 0, stream>>>(c0, c1, c2, counters);
    finalize_kernel<<<dim3(1), dim3(32), 0, stream>>>(desc, counters, (float*)d_out);
}